// GNNLayer_58935541236529
// MI455X (gfx1250) — compile-verified
//
#include <hip/hip_runtime.h>

#define H 128
#define EPS 1e-5f

typedef float v2f __attribute__((ext_vector_type(2)));
typedef float v8f __attribute__((ext_vector_type(8)));

// D(16x16,f32) = A(16x4,f32) x B(4x16,f32) + C   -- full fp32 precision WMMA
__device__ __forceinline__ v8f wmma_f32(v2f a, v2f b, v8f c) {
    return __builtin_amdgcn_wmma_f32_16x16x4_f32(
        /*neg_a=*/false, a, /*neg_b=*/false, b,
        /*c_mod=*/(short)0, c, /*reuse_a=*/false, /*reuse_b=*/false);
}

// ---------------------------------------------------------------------------
// Kernel 1: zero scratch (aggr + 4*H stat slots)
// ---------------------------------------------------------------------------
__global__ void zero_kernel(float* __restrict__ p, size_t n) {
    size_t i = (size_t)blockIdx.x * blockDim.x + threadIdx.x;
    size_t stride = (size_t)gridDim.x * blockDim.x;
    for (; i < n; i += stride) p[i] = 0.0f;
}

// ---------------------------------------------------------------------------
// Kernel 2: fused node GEMMs  yW = x @ W.T  for W1, W2, W4, W5.
// One wave per 16x16 output tile; A tile of x loaded once, 4 WMMA chains.
// ---------------------------------------------------------------------------
__global__ void node_gemm4_kernel(const float* __restrict__ x,
                                  const float* __restrict__ W1,
                                  const float* __restrict__ W2,
                                  const float* __restrict__ W4,
                                  const float* __restrict__ W5,
                                  float* __restrict__ xW1, float* __restrict__ xW2,
                                  float* __restrict__ xW4, float* __restrict__ xW5,
                                  int V) {
    const int lane  = threadIdx.x & 31;
    const int gw    = (int)((blockIdx.x * blockDim.x + threadIdx.x) >> 5);
    const int ntile = gw & 7;          // H/16 = 8 column tiles
    const int mtile = gw >> 3;
    const int m0 = mtile * 16;
    if (m0 >= V) return;               // wave-uniform: EXEC stays all-ones
    const int n0 = ntile * 16;

    // A: lane<16 holds rows m0+lane K={k0,k0+1}; lane>=16 same rows K={k0+2,k0+3}
    const int koff = (lane >> 4) * 2;
    const float* ap  = x  + (size_t)(m0 + (lane & 15)) * H + koff;
    // B[k][n] = W[n0+n][k0+k]  (y = x @ W.T); same per-lane addressing shape
    const float* b1p = W1 + (size_t)(n0 + (lane & 15)) * H + koff;
    const float* b2p = W2 + (size_t)(n0 + (lane & 15)) * H + koff;
    const float* b4p = W4 + (size_t)(n0 + (lane & 15)) * H + koff;
    const float* b5p = W5 + (size_t)(n0 + (lane & 15)) * H + koff;

    v8f acc1 = {}, acc2 = {}, acc4 = {}, acc5 = {};
#pragma unroll
    for (int k0 = 0; k0 < H; k0 += 4) {
        v2f a = *(const v2f*)(ap + k0);
        acc1 = wmma_f32(a, *(const v2f*)(b1p + k0), acc1);
        acc2 = wmma_f32(a, *(const v2f*)(b2p + k0), acc2);
        acc4 = wmma_f32(a, *(const v2f*)(b4p + k0), acc4);
        acc5 = wmma_f32(a, *(const v2f*)(b5p + k0), acc5);
    }

    // C/D: vgpr r, lanes0-15 -> row r, lanes16-31 -> row r+8; col = lane&15
    const int col = n0 + (lane & 15);
    const int rb  = (lane >> 4) * 8;
#pragma unroll
    for (int r = 0; r < 8; ++r) {
        size_t o = (size_t)(m0 + rb + r) * H + col;
        xW1[o] = acc1[r];
        xW2[o] = acc2[r];
        xW4[o] = acc4[r];
        xW5[o] = acc5[r];
    }
}

// ---------------------------------------------------------------------------
// Kernel 3: messages = e * xW2[src]; scatter-add into aggr[dest] (f32 atomics,
// target fits in L2). One wave per edge, lane handles 4 features (float4).
// ---------------------------------------------------------------------------
__global__ void scatter_kernel(const float* __restrict__ e,
                               const float* __restrict__ xW2,
                               const int* __restrict__ src,
                               const int* __restrict__ dest,
                               float* __restrict__ aggr, int E) {
    const int gw   = (int)((blockIdx.x * blockDim.x + threadIdx.x) >> 5);
    const int lane = threadIdx.x & 31;
    if (gw >= E) return;
    const int s = src[gw];
    const int d = dest[gw];
    const int h = lane * 4;
    const float4 ev = *(const float4*)(e   + (size_t)gw * H + h);
    const float4 wv = *(const float4*)(xW2 + (size_t)s  * H + h);
    float* base = aggr + (size_t)d * H + h;
    atomicAdd(base + 0, ev.x * wv.x);
    atomicAdd(base + 1, ev.y * wv.y);
    atomicAdd(base + 2, ev.z * wv.z);
    atomicAdd(base + 3, ev.w * wv.w);
}

// ---------------------------------------------------------------------------
// Kernel 4: node BN stats over pre_x = xW1 + aggr (column sum / sumsq)
// ---------------------------------------------------------------------------
__global__ void node_stats_kernel(const float* __restrict__ xW1,
                                  const float* __restrict__ aggr,
                                  float* __restrict__ nsum,
                                  float* __restrict__ nsumsq, int V) {
    const int c = threadIdx.x;   // blockDim.x == H
    float s = 0.0f, ss = 0.0f;
    for (int v = blockIdx.x; v < V; v += gridDim.x) {
        float val = xW1[(size_t)v * H + c] + aggr[(size_t)v * H + c];
        s  += val;
        ss += val * val;
    }
    atomicAdd(&nsum[c], s);
    atomicAdd(&nsumsq[c], ss);
}

// ---------------------------------------------------------------------------
// Kernel 5: node BN apply + relu + residual -> x_new
// ---------------------------------------------------------------------------
__global__ void node_apply_kernel(const float* __restrict__ x,
                                  const float* __restrict__ xW1,
                                  const float* __restrict__ aggr,
                                  const float* __restrict__ nsum,
                                  const float* __restrict__ nsumsq,
                                  const float* __restrict__ gamma_x,
                                  const float* __restrict__ beta_x,
                                  float* __restrict__ out_x,
                                  float invV, size_t total) {
    size_t i = (size_t)blockIdx.x * blockDim.x + threadIdx.x;
    if (i >= total) return;
    const int c = (int)(i & (H - 1));
    float mean = nsum[c] * invV;
    float var  = nsumsq[c] * invV - mean * mean;
    float val  = xW1[i] + aggr[i];
    float nrm  = (val - mean) * rsqrtf(var + EPS) * gamma_x[c] + beta_x[c];
    out_x[i] = x[i] + fmaxf(nrm, 0.0f);
}

// ---------------------------------------------------------------------------
// Edge tile GEMM helper: 16x16 tile of e @ W3.T via WMMA f32
// ---------------------------------------------------------------------------
__device__ __forceinline__ v8f edge_tile_gemm(const float* __restrict__ e,
                                              const float* __restrict__ W3,
                                              int m0, int n0, int lane) {
    const int koff = (lane >> 4) * 2;
    const float* ap = e  + (size_t)(m0 + (lane & 15)) * H + koff;
    const float* bp = W3 + (size_t)(n0 + (lane & 15)) * H + koff;
    v8f acc = {};
#pragma unroll
    for (int k0 = 0; k0 < H; k0 += 4)
        acc = wmma_f32(*(const v2f*)(ap + k0), *(const v2f*)(bp + k0), acc);
    return acc;
}

// ---------------------------------------------------------------------------
// Kernel 6: edge BN stats pass.  pre_e = e@W3.T + xW4[dest] + xW5[src] is
// computed per tile (not stored); per-lane stats accumulated over a strip of
// M-tiles, then one atomic per lane per stat.
// ---------------------------------------------------------------------------
__global__ void edge_stats_kernel(const float* __restrict__ e,
                                  const float* __restrict__ W3,
                                  const float* __restrict__ xW4,
                                  const float* __restrict__ xW5,
                                  const int* __restrict__ src,
                                  const int* __restrict__ dest,
                                  float* __restrict__ esum,
                                  float* __restrict__ esumsq,
                                  int Etiles, int striplen) {
    const int lane  = threadIdx.x & 31;
    const int gw    = (int)((blockIdx.x * blockDim.x + threadIdx.x) >> 5);
    const int ntile = gw & 7;
    const int strip = gw >> 3;
    const int t0 = strip * striplen;
    if (t0 >= Etiles) return;        // wave-uniform
    const int t1 = (t0 + striplen < Etiles) ? (t0 + striplen) : Etiles;
    const int n0  = ntile * 16;
    const int col = n0 + (lane & 15);
    const int rb  = (lane >> 4) * 8;

    float lsum = 0.0f, lsq = 0.0f;
    for (int t = t0; t < t1; ++t) {
        const int m0 = t * 16;
        v8f acc = edge_tile_gemm(e, W3, m0, n0, lane);
#pragma unroll
        for (int r = 0; r < 8; ++r) {
            const int row = m0 + rb + r;
            const int d = dest[row];
            const int s = src[row];
            float v = acc[r] + xW4[(size_t)d * H + col] + xW5[(size_t)s * H + col];
            lsum += v;
            lsq  += v * v;
        }
    }
    atomicAdd(&esum[col],   lsum);
    atomicAdd(&esumsq[col], lsq);
}

// ---------------------------------------------------------------------------
// Kernel 7: edge apply pass: recompute pre_e tile, BN + relu + residual -> e_new
// ---------------------------------------------------------------------------
__global__ void edge_apply_kernel(const float* __restrict__ e,
                                  const float* __restrict__ W3,
                                  const float* __restrict__ xW4,
                                  const float* __restrict__ xW5,
                                  const int* __restrict__ src,
                                  const int* __restrict__ dest,
                                  const float* __restrict__ esum,
                                  const float* __restrict__ esumsq,
                                  const float* __restrict__ gamma_e,
                                  const float* __restrict__ beta_e,
                                  float* __restrict__ out_e,
                                  float invE, int Etiles) {
    const int lane  = threadIdx.x & 31;
    const int gw    = (int)((blockIdx.x * blockDim.x + threadIdx.x) >> 5);
    const int ntile = gw & 7;
    const int mtile = gw >> 3;
    if (mtile >= Etiles) return;     // wave-uniform
    const int m0  = mtile * 16;
    const int n0  = ntile * 16;
    const int col = n0 + (lane & 15);
    const int rb  = (lane >> 4) * 8;

    const float mean = esum[col] * invE;
    const float var  = esumsq[col] * invE - mean * mean;
    const float rstd = rsqrtf(var + EPS);
    const float g = gamma_e[col];
    const float b = beta_e[col];

    v8f acc = edge_tile_gemm(e, W3, m0, n0, lane);
#pragma unroll
    for (int r = 0; r < 8; ++r) {
        const int row = m0 + rb + r;
        const int d = dest[row];
        const int s = src[row];
        float v   = acc[r] + xW4[(size_t)d * H + col] + xW5[(size_t)s * H + col];
        float nrm = (v - mean) * rstd * g + b;
        out_e[(size_t)row * H + col] = e[(size_t)row * H + col] + fmaxf(nrm, 0.0f);
    }
}

// ---------------------------------------------------------------------------
extern "C" void kernel_launch(void* const* d_in, const int* in_sizes, int n_in,
                              void* d_out, int out_size, void* d_ws, size_t ws_size,
                              hipStream_t stream) {
    const float* x       = (const float*)d_in[0];
    const float* e       = (const float*)d_in[1];
    const int*   eidx    = (const int*)  d_in[2];
    const float* W1      = (const float*)d_in[3];
    const float* W2      = (const float*)d_in[4];
    const float* W3      = (const float*)d_in[5];
    const float* W4      = (const float*)d_in[6];
    const float* W5      = (const float*)d_in[7];
    const float* gamma_x = (const float*)d_in[8];
    const float* beta_x  = (const float*)d_in[9];
    const float* gamma_e = (const float*)d_in[10];
    const float* beta_e  = (const float*)d_in[11];

    const int V = in_sizes[0] / H;
    const int E = in_sizes[1] / H;
    const int* src  = eidx;        // edge_index[0]
    const int* dest = eidx + E;    // edge_index[1]

    float* ws = (float*)d_ws;
    const size_t VH = (size_t)V * H;
    float* xW1  = ws;
    float* xW2  = ws + VH;
    float* xW4  = ws + 2 * VH;
    float* xW5  = ws + 3 * VH;
    float* aggr = ws + 4 * VH;
    float* nsum   = ws + 5 * VH;           // H
    float* nsumsq = nsum + H;              // H
    float* esum   = nsum + 2 * H;          // H
    float* esumsq = nsum + 3 * H;          // H

    float* out_x = (float*)d_out;
    float* out_e = out_x + VH;

    // 1) zero aggr + all stat slots (contiguous region)
    zero_kernel<<<1024, 256, 0, stream>>>(aggr, VH + 4 * (size_t)H);

    // 2) node GEMMs (xW1, xW2, xW4, xW5), one wave per 16x16 tile
    const int Vt = (V + 15) / 16;
    const int nodeWaves = Vt * 8;
    node_gemm4_kernel<<<(nodeWaves + 3) / 4, 128, 0, stream>>>(
        x, W1, W2, W4, W5, xW1, xW2, xW4, xW5, V);

    // 3) edge messages + scatter-add (one wave per edge)
    scatter_kernel<<<(E + 7) / 8, 256, 0, stream>>>(e, xW2, src, dest, aggr, E);

    // 4) node BN stats
    node_stats_kernel<<<512, H, 0, stream>>>(xW1, aggr, nsum, nsumsq, V);

    // 5) node BN apply -> x_new
    node_apply_kernel<<<(int)((VH + 255) / 256), 256, 0, stream>>>(
        x, xW1, aggr, nsum, nsumsq, gamma_x, beta_x, out_x, 1.0f / (float)V, VH);

    // 6) edge BN stats (strip pass, pre_e recomputed, not stored)
    const int Et = E / 16;
    const int striplen = 100;
    const int nstrips = (Et + striplen - 1) / striplen;
    edge_stats_kernel<<<(nstrips * 8 + 3) / 4, 128, 0, stream>>>(
        e, W3, xW4, xW5, src, dest, esum, esumsq, Et, striplen);

    // 7) edge apply -> e_new
    edge_apply_kernel<<<(Et * 8 + 3) / 4, 128, 0, stream>>>(
        e, W3, xW4, xW5, src, dest, esum, esumsq, gamma_e, beta_e,
        out_e, 1.0f / (float)E, Et);
}